// GNNBaseline_73778948210746
// MI455X (gfx1250) — compile-verified
//
#include <hip/hip_runtime.h>
#include <stdint.h>

// ---------------------------------------------------------------------------
// GAT forward (3 GATConv layers + output projection) for gfx1250 / MI455X.
// fp32 everywhere; dense GEMMs via V_WMMA_F32_16X16X4_F32 (wave32 WMMA).
// Edge softmax via encoded-uint atomicMax + float atomicAdd (L2 atomics).
// ---------------------------------------------------------------------------

#define GN 65536      // nodes
#define GE 1048576    // edges
#define WIDTH 64      // channel width after every GAT layer

typedef __attribute__((ext_vector_type(2))) float v2f;
typedef __attribute__((ext_vector_type(8))) float v8f;

// Monotonic float<->uint encoding for atomic max over floats.
__device__ __forceinline__ unsigned fenc(float x) {
    unsigned u = __float_as_uint(x);
    return (u & 0x80000000u) ? ~u : (u | 0x80000000u);
}
__device__ __forceinline__ float fdec(unsigned e) {
    return __uint_as_float((e & 0x80000000u) ? (e ^ 0x80000000u) : ~e);
}

// ---------------------------------------------------------------------------
// WMMA fp32 GEMM: Y[M,NOUT] = X[M,K] @ W[K,NOUT] (+ bias).
// One wave -> one 16x16 C tile, 8 waves per block.
// A frag (16x4 f32, 2 VGPR/lane): lanes 0-15 hold K={k,k+1}, lanes 16-31 K={k+2,k+3}.
// B frag (4x16 f32): VGPR0 = rows {k, k+2}, VGPR1 = rows {k+1, k+3} across lane halves.
// C/D (16x16 f32, 8 VGPR): VGPR r -> row r (lanes 0-15) / row r+8 (lanes 16-31).
// ---------------------------------------------------------------------------
template <int K, int NOUT, bool BIAS>
__global__ __launch_bounds__(256) void wmma_gemm_f32(
    const float* __restrict__ X, const float* __restrict__ W,
    const float* __restrict__ bias, float* __restrict__ Y)
{
    constexpr int COLT = NOUT / 16;   // column tiles covered by the 8 waves
    constexpr int ROWT = 8 / COLT;    // row tiles per block
    const int wave = threadIdx.x >> 5;
    const int lane = threadIdx.x & 31;
    const int colT = wave % COLT;
    const int rowT = blockIdx.x * ROWT + (wave / COLT);
    const int hi    = lane >> 4;          // 0: lanes 0-15, 1: lanes 16-31
    const int row   = rowT * 16 + (lane & 15);
    const int col   = colT * 16 + (lane & 15);
    const int khalf = hi * 2;

    const float* __restrict__ Xrow = X + (size_t)row * K;
    v8f c = {};
#pragma unroll 8
    for (int k = 0; k < K; k += 4) {
        v2f a, b;
        a.x = Xrow[k + khalf];
        a.y = Xrow[k + khalf + 1];
        b.x = W[(size_t)(k + khalf)     * NOUT + col];
        b.y = W[(size_t)(k + khalf + 1) * NOUT + col];
        c = __builtin_amdgcn_wmma_f32_16x16x4_f32(
                false, a, false, b, (short)0, c, false, false);
    }
    const int rbase = rowT * 16 + hi * 8;
#pragma unroll
    for (int r = 0; r < 8; ++r) {
        float v = c[r];
        if (BIAS) v += bias[col];
        Y[(size_t)(rbase + r) * NOUT + col] = v;
    }
}

// a_s[n,h] = xp[n,h,:]·a_src[h,:] ; a_d likewise. One thread per (node,head).
template <int H>
__global__ void node_attn(const float* __restrict__ xp,
                          const float* __restrict__ a_src,
                          const float* __restrict__ a_dst,
                          float* __restrict__ as_, float* __restrict__ ad_)
{
    int t = blockIdx.x * blockDim.x + threadIdx.x;
    if (t >= GN * H) return;
    constexpr int D = WIDTH / H;
    int h = t % H, n = t / H;
    const float* row = xp + (size_t)n * WIDTH + h * D;
    const float* s = a_src + h * D;
    const float* d = a_dst + h * D;
    float aS = 0.f, aD = 0.f;
#pragma unroll
    for (int k = 0; k < D; ++k) { float v = row[k]; aS += v * s[k]; aD += v * d[k]; }
    as_[t] = aS; ad_[t] = aD;
}

// ce[k,h] = sum_d We[k, h*D+d] * a_edge[h,d]  (16 x H result, single tiny block)
template <int H>
__global__ void edge_fold(const float* __restrict__ We,
                          const float* __restrict__ a_edge,
                          float* __restrict__ ce)
{
    int t = threadIdx.x;
    if (t >= 16 * H) return;
    constexpr int D = WIDTH / H;
    int k = t / H, h = t % H;
    float acc = 0.f;
#pragma unroll
    for (int d = 0; d < D; ++d) acc += We[(size_t)k * WIDTH + h * D + d] * a_edge[h * D + d];
    ce[k * H + h] = acc;
}

// Zero acc, denom; set amaxEnc to enc(-inf).
template <int H>
__global__ void init_layer(float* __restrict__ acc, unsigned* __restrict__ amaxEnc,
                           float* __restrict__ denom)
{
    int t = blockIdx.x * blockDim.x + threadIdx.x;
    if (t < GN * WIDTH) acc[t] = 0.f;
    if (t < GN * H) { amaxEnc[t] = 0x007FFFFFu; denom[t] = 0.f; }
}

// Pass A: alpha = leaky_relu(a_s[src] + a_d[dst] + a_e), atomic-max into amax[dst].
template <int H>
__global__ void edge_logits(const float* __restrict__ w, const int* __restrict__ ei,
                            const float* __restrict__ as_, const float* __restrict__ ad_,
                            const float* __restrict__ ce,
                            float* __restrict__ alpha, unsigned* __restrict__ amaxEnc)
{
    int e = blockIdx.x * blockDim.x + threadIdx.x;
    if (e >= GE) return;
    int s = ei[e], d = ei[GE + e];
    float wrow[16];
#pragma unroll
    for (int k = 0; k < 16; ++k) wrow[k] = w[(size_t)e * 16 + k];
#pragma unroll
    for (int h = 0; h < H; ++h) {
        float ae = 0.f;
#pragma unroll
        for (int k = 0; k < 16; ++k) ae += wrow[k] * ce[k * H + h];
        float al = as_[s * H + h] + ad_[d * H + h] + ae;
        al = (al > 0.f) ? al : 0.2f * al;           // leaky_relu, slope 0.2
        alpha[(size_t)e * H + h] = al;
        atomicMax(&amaxEnc[d * H + h], fenc(al));
    }
}

// amax = where(isfinite(amax), amax, 0)
template <int H>
__global__ void amax_fix(const unsigned* __restrict__ amaxEnc, float* __restrict__ amax)
{
    int t = blockIdx.x * blockDim.x + threadIdx.x;
    if (t >= GN * H) return;
    float v = fdec(amaxEnc[t]);
    amax[t] = (v > -3.0e38f && v < 3.0e38f) ? v : 0.f;
}

// Pass B: ex = exp(alpha - amax[dst]); denom[dst] += ex  (alpha buffer overwritten)
template <int H>
__global__ void edge_exp(const int* __restrict__ ei, const float* __restrict__ amax,
                         float* __restrict__ alpha_ex, float* __restrict__ denom)
{
    int e = blockIdx.x * blockDim.x + threadIdx.x;
    if (e >= GE) return;
    int d = ei[GE + e];
#pragma unroll
    for (int h = 0; h < H; ++h) {
        float ex = __expf(alpha_ex[(size_t)e * H + h] - amax[d * H + h]);
        alpha_ex[(size_t)e * H + h] = ex;
        atomicAdd(&denom[d * H + h], ex);
    }
}

// Pass C: acc[dst, c] += xp[src, c] * att. Thread = (edge, 4-channel group).
template <int H>
__global__ void edge_message(const int* __restrict__ ei, const float* __restrict__ xp,
                             const float* __restrict__ ex, const float* __restrict__ denom,
                             float* __restrict__ acc)
{
    long long t = (long long)blockIdx.x * blockDim.x + threadIdx.x;
    if (t >= (long long)GE * 16) return;
    int e = (int)(t >> 4);
    int c = (int)(t & 15) * 4;
    constexpr int D = WIDTH / H;
    int h = c / D;                                  // D is 32 or 64; groups never split heads
    int s = ei[e], d = ei[GE + e];
    float att = ex[(size_t)e * H + h] / (denom[d * H + h] + 1e-16f);
    const float4 xv = *(const float4*)(xp + (size_t)s * WIDTH + c);
    float* dst = acc + (size_t)d * WIDTH + c;
    atomicAdd(dst + 0, xv.x * att);
    atomicAdd(dst + 1, xv.y * att);
    atomicAdd(dst + 2, xv.z * att);
    atomicAdd(dst + 3, xv.w * att);
}

// x = silu(acc + b)
__global__ void bias_silu(const float* __restrict__ acc, const float* __restrict__ b,
                          float* __restrict__ x)
{
    int t = blockIdx.x * blockDim.x + threadIdx.x;
    if (t >= GN * WIDTH) return;
    float v = acc[t] + b[t & (WIDTH - 1)];
    x[t] = v / (1.f + __expf(-v));
}

// ---------------------------------------------------------------------------
extern "C" void kernel_launch(void* const* d_in, const int* in_sizes, int n_in,
                              void* d_out, int out_size, void* d_ws, size_t ws_size,
                              hipStream_t stream)
{
    (void)in_sizes; (void)n_in; (void)out_size; (void)ws_size;

    const float* h_in = (const float*)d_in[0];            // [N,128]
    const float* w_in = (const float*)d_in[1];            // [E,16]
    // params[i]: W, We, a_src, a_dst, a_edge, b at indices 2+6i .. 7+6i
    const float* Wout = (const float*)d_in[20];           // [64,128]
    const float* bout = (const float*)d_in[21];           // [128]
    const int*   ei   = (const int*)d_in[22];             // [2,E]
    float* out = (float*)d_out;                           // [N,128]

    // Workspace carve-up
    char* ws = (char*)d_ws;
    size_t off = 0;
    auto carve = [&](size_t bytes) { void* p = ws + off; off = (off + bytes + 255) & ~(size_t)255; return p; };
    float*    xp      = (float*)   carve((size_t)GN * WIDTH * 4);  // node features post-W
    float*    acc     = (float*)   carve((size_t)GN * WIDTH * 4);  // scatter accumulator
    float*    xbuf    = (float*)   carve((size_t)GN * WIDTH * 4);  // layer activations
    float*    alpha   = (float*)   carve((size_t)GE * 2 * 4);      // logits -> exp
    float*    as_     = (float*)   carve((size_t)GN * 2 * 4);
    float*    ad_     = (float*)   carve((size_t)GN * 2 * 4);
    unsigned* amaxEnc = (unsigned*)carve((size_t)GN * 2 * 4);
    float*    amaxF   = (float*)   carve((size_t)GN * 2 * 4);
    float*    denom   = (float*)   carve((size_t)GN * 2 * 4);
    float*    ce      = (float*)   carve(256);

    const int TB = 256;
    const int nwH2   = (GN * 2 + TB - 1) / TB;
    const int nwH1   = (GN * 1 + TB - 1) / TB;
    const int nwFull = (GN * WIDTH + TB - 1) / TB;
    const int eBlk   = (GE + TB - 1) / TB;
    const int mBlk   = (int)(((long long)GE * 16 + TB - 1) / TB);

    // ---------------- Layer 0: in=128, heads=2 ----------------
    {
        const float* W  = (const float*)d_in[2];
        const float* We = (const float*)d_in[3];
        const float* aS = (const float*)d_in[4];
        const float* aD = (const float*)d_in[5];
        const float* aE = (const float*)d_in[6];
        const float* b  = (const float*)d_in[7];

        wmma_gemm_f32<128, 64, false><<<GN / 32, 256, 0, stream>>>(h_in, W, nullptr, xp);
        node_attn<2><<<nwH2, TB, 0, stream>>>(xp, aS, aD, as_, ad_);
        edge_fold<2><<<1, 32, 0, stream>>>(We, aE, ce);
        init_layer<2><<<nwFull, TB, 0, stream>>>(acc, amaxEnc, denom);
        edge_logits<2><<<eBlk, TB, 0, stream>>>(w_in, ei, as_, ad_, ce, alpha, amaxEnc);
        amax_fix<2><<<nwH2, TB, 0, stream>>>(amaxEnc, amaxF);
        edge_exp<2><<<eBlk, TB, 0, stream>>>(ei, amaxF, alpha, denom);
        edge_message<2><<<mBlk, TB, 0, stream>>>(ei, xp, alpha, denom, acc);
        bias_silu<<<nwFull, TB, 0, stream>>>(acc, b, xbuf);
    }

    // ---------------- Layers 1,2: in=64, heads=1 ----------------
    for (int L = 1; L <= 2; ++L) {
        const float* W  = (const float*)d_in[2 + 6 * L];
        const float* We = (const float*)d_in[3 + 6 * L];
        const float* aS = (const float*)d_in[4 + 6 * L];
        const float* aD = (const float*)d_in[5 + 6 * L];
        const float* aE = (const float*)d_in[6 + 6 * L];
        const float* b  = (const float*)d_in[7 + 6 * L];

        wmma_gemm_f32<64, 64, false><<<GN / 32, 256, 0, stream>>>(xbuf, W, nullptr, xp);
        node_attn<1><<<nwH1, TB, 0, stream>>>(xp, aS, aD, as_, ad_);
        edge_fold<1><<<1, 32, 0, stream>>>(We, aE, ce);
        init_layer<1><<<nwFull, TB, 0, stream>>>(acc, amaxEnc, denom);
        edge_logits<1><<<eBlk, TB, 0, stream>>>(w_in, ei, as_, ad_, ce, alpha, amaxEnc);
        amax_fix<1><<<nwH1, TB, 0, stream>>>(amaxEnc, amaxF);
        edge_exp<1><<<eBlk, TB, 0, stream>>>(ei, amaxF, alpha, denom);
        edge_message<1><<<mBlk, TB, 0, stream>>>(ei, xp, alpha, denom, acc);
        bias_silu<<<nwFull, TB, 0, stream>>>(acc, b, xbuf);
    }

    // ---------------- Output projection: [N,64]@[64,128]+b ----------------
    wmma_gemm_f32<64, 128, true><<<GN / 16, 256, 0, stream>>>(xbuf, Wout, bout, out);
}